// SoftmaxMatcher_10058813407693
// MI455X (gfx1250) — compile-verified
//
#include <hip/hip_runtime.h>

typedef __attribute__((ext_vector_type(2))) float v2f;
typedef __attribute__((ext_vector_type(8))) float v8f;

#define C_DIM 256
#define N_PTS 1024
#define H_DIM 256
#define W_DIM 256
#define HW_DIM (H_DIM * W_DIM)
#define KEEP 13
#define WMAX 27
#define TEMP_INV 100.0f
#define EPS_N 1e-12f
#define SU_TOTAL 8355840.0f   // H * sum_{x=0}^{W-1} x = 256 * 32640 (exact in f32)

// ---------------------------------------------------------------------------
// Kernel 1: per-pixel inverse L2 norm of target dense descriptors (batches 1,3)
// Coalesced: consecutive threads read consecutive pixels within a channel.
// ---------------------------------------------------------------------------
__global__ void tgt_invnorm_kernel(const float* __restrict__ dd, float* __restrict__ invn) {
    int gid = blockIdx.x * blockDim.x + threadIdx.x;   // 0 .. 2*HW-1
    int b   = gid >> 16;
    int pix = gid & (HW_DIM - 1);
    int db  = 2 * b + 1;
    const float* p = dd + (size_t)db * C_DIM * HW_DIM + pix;
    float s = 0.f;
    #pragma unroll 4
    for (int c = 0; c < C_DIM; ++c) {
        float v = p[(size_t)c * HW_DIM];
        s += v * v;
    }
    invn[gid] = 1.0f / fmaxf(sqrtf(s), EPS_N);
}

// ---------------------------------------------------------------------------
// Kernel 2: normalize src keypoint descriptors, store keypoint-contiguous
// srcn[(b*N + n)*C + c]
// ---------------------------------------------------------------------------
__global__ void src_norm_kernel(const float* __restrict__ kd, float* __restrict__ srcn) {
    __shared__ float red[256];
    int blk = blockIdx.x;        // 0..2047  (b*1024 + n)
    int b   = blk >> 10;
    int n   = blk & 1023;
    int kb  = 2 * b;             // kp_inds = [0, 2]
    int c   = threadIdx.x;
    float v = kd[(size_t)kb * C_DIM * N_PTS + (size_t)c * N_PTS + n];
    red[c] = v * v;
    __syncthreads();
    for (int s = 128; s > 0; s >>= 1) {
        if (c < s) red[c] += red[c + s];
        __syncthreads();
    }
    float inv = 1.0f / fmaxf(sqrtf(red[0]), EPS_N);
    srcn[(size_t)blk * C_DIM + c] = v * inv;
}

// ---------------------------------------------------------------------------
// Kernel 3: per-keypoint windowed correlation (f32 WMMA), exact masked softmax
// with analytic out-of-window tail, soft-argmax, bilinear sampling, weights.
// One 256-thread block (8 waves) per keypoint.
// ---------------------------------------------------------------------------
__global__ void matcher_kernel(const float* __restrict__ kp_scores,
                               const float* __restrict__ scores_dense,
                               const float* __restrict__ desc_dense,
                               const float* __restrict__ coords,
                               const float* __restrict__ srcn,
                               const float* __restrict__ invn,
                               float* __restrict__ out) {
    __shared__ float srcn_s[C_DIM];
    __shared__ float corr_s[WMAX * WMAX];
    __shared__ float redA[256];
    __shared__ float redB[256];
    __shared__ float redC[256];

    const int tid = threadIdx.x;
    const int blk = blockIdx.x;        // b*1024 + n
    const int b   = blk >> 10;
    const int n   = blk & 1023;
    const int kb  = 2 * b;             // src batch (0,2)
    const int db  = 2 * b + 1;         // dense batch (1,3)

    srcn_s[tid] = srcn[(size_t)blk * C_DIM + tid];

    const float u = coords[((size_t)kb * N_PTS + n) * 2 + 0];
    const float v = coords[((size_t)kb * N_PTS + n) * 2 + 1];
    const int ui = (int)u;   // trunc (coords >= 0)
    const int vi = (int)v;
    const int ulo = max(ui - KEEP, 0), uhi = min(ui + KEEP, W_DIM - 1);
    const int vlo = max(vi - KEEP, 0), vhi = min(vi + KEEP, H_DIM - 1);
    const int wlen = uhi - ulo + 1, hlen = vhi - vlo + 1;
    const int P = wlen * hlen;
    const int ngroups = (wlen + 15) >> 4;
    const int T = hlen * ngroups;

    __syncthreads();

    const float* dbase = desc_dense + (size_t)db * C_DIM * HW_DIM;
    const float* ibase = invn + (size_t)b * HW_DIM;

    const int wid  = tid >> 5;
    const int lane = tid & 31;
    const int half = lane >> 4;     // selects K-pair (0,1) vs (2,3)
    const int mcol = lane & 15;     // M index (pixel within 16-group)

    // ---- windowed correlations via V_WMMA_F32_16X16X4_F32 ----
    for (int t = wid; t < T; t += 8) {
        const int rowrel = t / ngroups;
        const int g = t - rowrel * ngroups;
        const int y = vlo + rowrel;
        const int xg = ulo + (g << 4);
        const int x = xg + mcol;
        const int xc = min(x, W_DIM - 1);          // clamp loads; invalid M ignored
        const float* pixp = dbase + (size_t)y * W_DIM + xc;
        const float inv = ibase[y * W_DIM + xc];   // per-pixel target inv-norm folded into A
        v8f acc = {};
        #pragma unroll 8
        for (int k = 0; k < C_DIM; k += 4) {
            const int c0 = k + 2 * half;
            v2f a, bb;
            a[0]  = pixp[(size_t)c0 * HW_DIM] * inv;
            a[1]  = pixp[(size_t)(c0 + 1) * HW_DIM] * inv;
            bb[0] = srcn_s[c0];         // B replicated across all 16 columns
            bb[1] = srcn_s[c0 + 1];
            acc = __builtin_amdgcn_wmma_f32_16x16x4_f32(
                false, a, false, bb, (short)0, acc, false, false);
        }
        // D: VGPR r = row M=r (lanes 0-15) / M=r+8 (lanes 16-31); all columns equal
        if (mcol == 0) {
            const int base = (g << 4) + (half << 3);
            #pragma unroll
            for (int r = 0; r < 8; ++r) {
                const int col = base + r;
                if (col < wlen) corr_s[rowrel * wlen + col] = acc[r];
            }
        }
    }
    __syncthreads();

    // ---- max (out-of-window masked entries are exactly 0.0) ----
    float lm = 0.0f;
    for (int p = tid; p < P; p += 256) lm = fmaxf(lm, corr_s[p]);
    redA[tid] = lm;
    __syncthreads();
    for (int s = 128; s > 0; s >>= 1) {
        if (tid < s) redA[tid] = fmaxf(redA[tid], redA[tid + s]);
        __syncthreads();
    }
    const float m = redA[0];
    __syncthreads();

    // ---- in-window exp sums and coordinate moments ----
    float ls = 0.f, lsu = 0.f, lsv = 0.f;
    for (int p = tid; p < P; p += 256) {
        const float e = __expf((corr_s[p] - m) * TEMP_INV);
        const int row = p / wlen;
        const int col = p - row * wlen;
        ls += e;
        lsu += e * (float)(ulo + col);
        lsv += e * (float)(vlo + row);
    }
    redA[tid] = ls; redB[tid] = lsu; redC[tid] = lsv;
    __syncthreads();
    for (int s = 128; s > 0; s >>= 1) {
        if (tid < s) {
            redA[tid] += redA[tid + s];
            redB[tid] += redB[tid + s];
            redC[tid] += redC[tid + s];
        }
        __syncthreads();
    }
    const float S = redA[0], SumU = redB[0], SumV = redC[0];
    __syncthreads();

    // ---- analytic out-of-window tail: (HW-P) cells of value 0 ----
    const float z = __expf(-m * TEMP_INV);
    const float sumWinU = (float)hlen * 0.5f * (float)(ulo + uhi) * (float)wlen;
    const float sumWinV = (float)wlen * 0.5f * (float)(vlo + vhi) * (float)hlen;
    const float denom = S + (float)(HW_DIM - P) * z;
    const float pu = (SumU + z * (SU_TOTAL - sumWinU)) / denom;
    const float pv = (SumV + z * (SU_TOTAL - sumWinV)) / denom;

    // ---- bilinear sample setup (align_corners=True norm -> AC=False sample) ----
    const float xs = pu * ((float)W_DIM / (float)(W_DIM - 1)) - 0.5f;
    const float ys = pv * ((float)H_DIM / (float)(H_DIM - 1)) - 0.5f;
    const float x0f = floorf(xs), y0f = floorf(ys);
    const float fx = xs - x0f, fy = ys - y0f;
    const int ix0 = (int)x0f, iy0 = (int)y0f;
    const int ix1 = ix0 + 1, iy1 = iy0 + 1;
    const float vx0 = (ix0 >= 0 && ix0 < W_DIM) ? 1.f : 0.f;
    const float vx1 = (ix1 >= 0 && ix1 < W_DIM) ? 1.f : 0.f;
    const float vy0 = (iy0 >= 0 && iy0 < H_DIM) ? 1.f : 0.f;
    const float vy1 = (iy1 >= 0 && iy1 < H_DIM) ? 1.f : 0.f;
    const int cx0 = min(max(ix0, 0), W_DIM - 1), cx1 = min(max(ix1, 0), W_DIM - 1);
    const int cy0 = min(max(iy0, 0), H_DIM - 1), cy1 = min(max(iy1, 0), H_DIM - 1);
    const int p00 = cy0 * W_DIM + cx0, p10 = cy0 * W_DIM + cx1;
    const int p01 = cy1 * W_DIM + cx0, p11 = cy1 * W_DIM + cx1;
    const float w00 = (1.f - fx) * (1.f - fy) * vx0 * vy0;
    const float w10 = fx * (1.f - fy) * vx1 * vy0;
    const float w01 = (1.f - fx) * fy * vx0 * vy1;
    const float w11 = fx * fy * vx1 * vy1;

    // ---- dot(srcn, bilinear-sampled raw desc): one channel per thread ----
    const float* chp = dbase + (size_t)tid * HW_DIM;
    const float pd = w00 * chp[p00] + w10 * chp[p10] + w01 * chp[p01] + w11 * chp[p11];
    redA[tid] = srcn_s[tid] * pd;
    __syncthreads();
    for (int s = 128; s > 0; s >>= 1) {
        if (tid < s) redA[tid] += redA[tid + s];
        __syncthreads();
    }

    if (tid == 0) {
        const float* sb = scores_dense + (size_t)db * HW_DIM;
        const float ps = w00 * sb[p00] + w10 * sb[p10] + w01 * sb[p01] + w11 * sb[p11];
        const float ks = kp_scores[(size_t)kb * N_PTS + n];
        const float mw = 0.5f * (redA[0] * (1.0f / (float)C_DIM) + 1.0f) * ks * ps;
        const size_t oi = (size_t)b * N_PTS + n;
        out[oi * 2 + 0] = pu;                       // pseudo_coords [2,1024,2]
        out[oi * 2 + 1] = pv;
        out[2 * (size_t)N_PTS * 2 + oi] = mw;       // match_weights [2,1,1024]
        if (blk == 0) {                             // kp_inds = [0, 2]
            out[6144] = 0.0f;
            out[6145] = 2.0f;
        }
    }
}

// ---------------------------------------------------------------------------
extern "C" void kernel_launch(void* const* d_in, const int* in_sizes, int n_in,
                              void* d_out, int out_size, void* d_ws, size_t ws_size,
                              hipStream_t stream) {
    (void)in_sizes; (void)n_in; (void)out_size; (void)ws_size;
    const float* kp_scores = (const float*)d_in[0];   // [4,1,1024]
    const float* kp_desc   = (const float*)d_in[1];   // [4,256,1024]
    const float* scores_d  = (const float*)d_in[2];   // [4,1,256,256]
    const float* desc_d    = (const float*)d_in[3];   // [4,256,256,256]
    const float* coords    = (const float*)d_in[4];   // [4,1024,2]
    float* out = (float*)d_out;
    float* ws  = (float*)d_ws;
    float* invn = ws;                 // 2*65536 floats
    float* srcn = ws + 2 * HW_DIM;    // 2*1024*256 floats

    tgt_invnorm_kernel<<<512, 256, 0, stream>>>(desc_d, invn);
    src_norm_kernel<<<2048, 256, 0, stream>>>(kp_desc, srcn);
    matcher_kernel<<<2048, 256, 0, stream>>>(kp_scores, scores_d, desc_d, coords,
                                             srcn, invn, out);
}